// BatchRankingLoss_32341103739407
// MI455X (gfx1250) — compile-verified
//
#include <hip/hip_runtime.h>
#include <hip/hip_bf16.h>

typedef __attribute__((ext_vector_type(2))) float v2f;
typedef __attribute__((ext_vector_type(8))) float v8f;

#define GAP 1.0f
#define THRESHOLD 0.1f

// ---------------------------------------------------------------------------
// Zero the scalar output (harness poisons d_out with 0xAA before timing).
// ---------------------------------------------------------------------------
__global__ void brl_zero_kernel(float* out) {
    if (threadIdx.x == 0 && blockIdx.x == 0) out[0] = 0.0f;
}

// ---------------------------------------------------------------------------
// Main pairwise hinge kernel.
//
// Tile decomposition: 16x16 (i,j) tiles. Each wave32 owns one i-tile and
// loops over a grid-strided set of j-tiles. For each tile pair we build two
// rank-2 WMMA products with V_WMMA_F32_16X16X4_F32:
//   P[m][n] = o_i[m] - o_j[n]   (A = [o_i, 1, 0, 0], B = [1; -o_j; 0; 0])
//   Q[m][n] = t_i[m] - t_j[n]   (A = [t_i, 1, 0, 0], B = [1; -t_j; 0; 0])
// A-layout (32-bit 16x4): lanes 0-15 hold K=0 (vgpr0) / K=1 (vgpr1),
// lanes 16-31 hold K=2/K=3 -> we zero those so only K=0,1 contribute.
// C/D layout (32-bit 16x16): lane L<16 -> N=L, M=vgpr; lane L>=16 -> N=L-16,
// M=vgpr+8. The hinge is applied element-wise on the accumulator registers.
// ---------------------------------------------------------------------------
__global__ void __launch_bounds__(256)
brl_pairwise_kernel(const float* __restrict__ O,   // predicted scores o[B]
                    const float* __restrict__ T,   // labels gdt_ts[B]
                    float* __restrict__ out,       // single f32 accumulator
                    int B, int nTiles, float invN) {
    const int lane   = threadIdx.x & 31;
    const int waveId = threadIdx.x >> 5;

    const int it = blockIdx.x * 8 + waveId;   // i-tile index for this wave
    if (it >= nTiles) return;
    const int i0 = it * 16;

    // Loop-invariant A operands: lanes 0-15 carry (value, 1.0); lanes 16-31
    // carry zeros so the K=2,3 positions of the 16x4 A matrix are zero.
    float oi = 0.0f, ti = 0.0f, oneA = 0.0f;
    if (lane < 16) {
        const int r = i0 + lane;
        if (r < B) { oi = O[r]; ti = T[r]; oneA = 1.0f; }
    }
    const v2f a_p = { oi, oneA };   // A for P: [o_i | 1]
    const v2f a_q = { ti, oneA };   // A for Q: [t_i | 1]

    float lsum = 0.0f;

    for (int jt = (int)blockIdx.y; jt < nTiles; jt += (int)gridDim.y) {
        const int j0 = jt * 16;

        // B operands: row K=0 = ones, row K=1 = -o_j / -t_j (lanes 0-15);
        // lanes 16-31 (K=2,3) zeroed.
        float oj = 0.0f, tj = 0.0f, oneB = 0.0f;
        if (lane < 16) {
            const int c = j0 + lane;
            if (c < B) { oj = O[c]; tj = T[c]; oneB = 1.0f; }
        }

        // Prefetch next j-tile (speculative, L2-resident anyway).
        {
            const int jn = jt + (int)gridDim.y;
            if (jn < nTiles && lane < 16) {
                __builtin_prefetch(&O[jn * 16 + lane], 0, 0);
                __builtin_prefetch(&T[jn * 16 + lane], 0, 0);
            }
        }

        const v2f b_p = { oneB, -oj };
        const v2f b_q = { oneB, -tj };

        v8f zc = {};
        // P[m][n] = o_i[m] - o_j[n]
        v8f P = __builtin_amdgcn_wmma_f32_16x16x4_f32(
            false, a_p, false, b_p, (short)0, zc, false, false);
        // Q[m][n] = t_i[m] - t_j[n]
        v8f Q = __builtin_amdgcn_wmma_f32_16x16x4_f32(
            false, a_q, false, b_q, (short)0, zc, false, false);

        // Hinge on the 8 accumulator elements this lane owns.
#pragma unroll
        for (int v = 0; v < 8; ++v) {
            const float p = P[v];
            const float q = Q[v];
            // y = -1 if (t_i - t_j) < 0 else +1  ->  y*p
            const float yp = (q < 0.0f) ? -p : p;
            const float h  = fmaxf(0.0f, GAP + yp);
            // weight: |t_i - t_j| > threshold (diagonal q==0 drops out)
            if (fabsf(q) > THRESHOLD) lsum += h;
        }
    }

    // Wave32 reduction, then one atomic per wave.
    float s = lsum;
#pragma unroll
    for (int off = 16; off > 0; off >>= 1) s += __shfl_xor(s, off, 32);
    if (lane == 0) atomicAdd(out, s * invN);
}

// ---------------------------------------------------------------------------
// Host launcher.
// ---------------------------------------------------------------------------
extern "C" void kernel_launch(void* const* d_in, const int* in_sizes, int n_in,
                              void* d_out, int out_size, void* d_ws, size_t ws_size,
                              hipStream_t stream) {
    const float* O = (const float*)d_in[0];   // input (B,1) f32 -> o[B]
    const float* T = (const float*)d_in[1];   // gdt_ts (B,) f32
    float* out = (float*)d_out;               // loss, shape [1], f32

    const int B = in_sizes[1];
    const int nTiles = (B + 15) / 16;
    const double N = (double)B * (double)(B - 1);
    const float invN = (float)(1.0 / N);

    brl_zero_kernel<<<1, 1, 0, stream>>>(out);

    dim3 block(256, 1, 1);                      // 8 wave32 per block
    dim3 grid((nTiles + 7) / 8, 16, 1);         // i-groups x j-slices
    brl_pairwise_kernel<<<grid, block, 0, stream>>>(O, T, out, B, nTiles, invN);
}